// ModelTree_50190987821777
// MI455X (gfx1250) — compile-verified
//
#include <hip/hip_runtime.h>

#define NSAMP  64
#define NCELL  50000
#define NFEAT  10
#define NNODE  15
#define NLEAF  8
#define CB     25      // cell-blocks per sample (grid.y)
#define TPB    256     // 8 waves of 32
#define C_K    14.4269504088896341f   // LOGISTIC_K / ln(2)

typedef __attribute__((ext_vector_type(2))) float v2f;
typedef __attribute__((ext_vector_type(8))) float v8f;

__device__ __forceinline__ float logsig(float z) {
    // stable log_sigmoid for the tiny finalize kernel only
    return fminf(z, 0.0f) - log1pf(__expf(-fabsf(z)));
}
__device__ __forceinline__ float sigmoidf(float z) {
    return 1.0f / (1.0f + __expf(-z));
}

__global__ void zero_acc_kernel(float* acc) {
    acc[threadIdx.x] = 0.0f;   // 512 floats
}

__global__ __launch_bounds__(TPB)
void tree_partial_kernel(const float* __restrict__ x,
                         const float* __restrict__ gate_params,
                         const int*   __restrict__ d1,
                         const int*   __restrict__ d2,
                         const float* __restrict__ path_mat,
                         float*       __restrict__ acc) {
    const int s    = blockIdx.x;          // sample
    const int cbid = blockIdx.y;          // cell block
    const int lane = threadIdx.x & 31;
    const int warp = threadIdx.x >> 5;
    const int h    = lane >> 4;           // half-wave: K-pair selector
    const int col  = lane & 15;           // cell column within 16-cell group

    __shared__ float sAcc[NLEAF];
    if (threadIdx.x < NLEAF) sAcc[threadIdx.x] = 0.0f;
    __syncthreads();

    // ---- per-lane constants: node slots (k,j) -> node index 4k + 2h + j ----
    // Gates pre-scaled by C_K so the logistic terms become native exp2:
    //   t1 = exp2(a0 - C*x1), t2 = exp2(C*x1 - a1),
    //   t3 = exp2(a2 - C*x2), t4 = exp2(C*x2 - a3)
    // node log2-value  L = log2((1+t1)(1+t2)(1+t3)(1+t4))  (positive)
    int   nd1[8], nd2[8];
    float a0[8], a1[8], a2[8], a3[8];
    v2f   amat[4];
    #pragma unroll
    for (int k = 0; k < 4; ++k) {
        #pragma unroll
        for (int j = 0; j < 2; ++j) {
            const int n    = 4 * k + 2 * h + j;
            const int slot = 2 * k + j;
            if (n < NNODE) {
                nd1[slot] = d1[n];
                nd2[slot] = d2[n];
                a0[slot] = C_K * sigmoidf(gate_params[n * 4 + 0]);
                a1[slot] = C_K * sigmoidf(gate_params[n * 4 + 1]);
                a2[slot] = C_K * sigmoidf(gate_params[n * 4 + 2]);
                a3[slot] = C_K * sigmoidf(gate_params[n * 4 + 3]);
            } else {            // node 15 padding (A column is zero anyway)
                nd1[slot] = 0; nd2[slot] = 0;
                a0[slot] = 0.f; a1[slot] = 0.f; a2[slot] = 0.f; a3[slot] = 0.f;
            }
        }
        // A-matrix (16x4 f32) chunk k: lane L holds A[m][2h], A[m][2h+1], m = col
        float av0 = 0.0f, av1 = 0.0f;
        const int n0 = 4 * k + 2 * h;
        const int n1 = n0 + 1;
        if (col < NLEAF) {
            if (n0 < NNODE) av0 = path_mat[col * NNODE + n0];
            if (n1 < NNODE) av1 = path_mat[col * NNODE + n1];
        }
        amat[k].x = av0; amat[k].y = av1;
    }

    float accv[8];
    #pragma unroll
    for (int r = 0; r < 8; ++r) accv[r] = 0.0f;

    // block-uniform sample base -> SGPR base + 32-bit VGPR offsets for gathers
    const float* __restrict__ xs = x + (size_t)s * (NCELL * NFEAT);
    const int nGroups = NCELL / 16;       // 3125, exact
    const int stride  = CB * 8;           // waves per sample

    for (int cg = cbid * 8 + warp; cg < nGroups; cg += stride) {
        const int base = (cg * 16 + col) * NFEAT;       // 32-bit element offset
        // prefetch next strided group (speculative; tail overshoot is dropped)
        __builtin_prefetch(&xs[base + stride * 16 * NFEAT], 0, 3);

        v8f cc = {0.f, 0.f, 0.f, 0.f, 0.f, 0.f, 0.f, 0.f};
        #pragma unroll
        for (int k = 0; k < 4; ++k) {
            float lp[2];
            #pragma unroll
            for (int j = 0; j < 2; ++j) {
                const int slot = 2 * k + j;
                const float cx1 = C_K * xs[base + nd1[slot]];
                const float cx2 = C_K * xs[base + nd2[slot]];
                const float t1 = __builtin_amdgcn_exp2f(a0[slot] - cx1);
                const float t2 = __builtin_amdgcn_exp2f(cx1 - a1[slot]);
                const float t3 = __builtin_amdgcn_exp2f(a2[slot] - cx2);
                const float t4 = __builtin_amdgcn_exp2f(cx2 - a3[slot]);
                const float p  = ((1.0f + t1) * (1.0f + t2))
                               * ((1.0f + t3) * (1.0f + t4));
                lp[j] = __builtin_amdgcn_logf(p);       // +log2 of node product
            }
            v2f b; b.x = lp[0]; b.y = lp[1];
            // D = A(16x4,f32) x B(4x16,f32) + C  -- exact f32 path-sum einsum
            cc = __builtin_amdgcn_wmma_f32_16x16x4_f32(
                     false, amat[k], false, b, (short)0, cc, false, false);
        }
        // lanes 0-15 hold +Σpath log2 for [leaf r][cell col]; leaf prob = 2^-D
        if (h == 0) {
            #pragma unroll
            for (int r = 0; r < 8; ++r)
                accv[r] += __builtin_amdgcn_exp2f(-cc[r]);  // neg folds into modifier
        }
    }

    // wave butterfly reduce (upper-half lanes contributed exact zeros)
    #pragma unroll
    for (int r = 0; r < 8; ++r) {
        float t = accv[r];
        for (int off = 16; off > 0; off >>= 1) t += __shfl_xor(t, off, 32);
        if (lane == 0) atomicAdd(&sAcc[r], t);
    }
    __syncthreads();
    if (threadIdx.x < NLEAF)
        atomicAdd(&acc[s * NLEAF + threadIdx.x], sAcc[threadIdx.x]);
}

__global__ void finalize_kernel(const float* __restrict__ acc,
                                const float* __restrict__ y,
                                const float* __restrict__ gate_params,
                                const float* __restrict__ ref_gates,
                                const float* __restrict__ linear_w,
                                const float* __restrict__ linear_b,
                                float*       __restrict__ out) {
    __shared__ float sB[NSAMP];
    const int s = threadIdx.x;            // 64 threads

    float logit = linear_b[0];
    #pragma unroll
    for (int l = 0; l < NLEAF; ++l) {
        const float lp = acc[s * NLEAF + l] * (1.0f / (float)NCELL);
        out[s * NLEAF + l] = lp;                       // leaf_probs
        logit += lp * linear_w[l];
    }
    out[NSAMP * NLEAF + s] = sigmoidf(logit);          // y_pred

    const float ys = y[s];
    sB[s] = -(ys * logsig(logit) + (1.0f - ys) * logsig(-logit));
    __syncthreads();

    if (s == 0) {
        float bce = 0.0f;
        for (int i = 0; i < NSAMP; ++i) bce += sB[i];
        bce *= (1.0f / (float)NSAMP);
        float reg = 0.0f;
        for (int i = 0; i < NNODE * 4; ++i) {
            const float d = sigmoidf(gate_params[i]) - ref_gates[i];
            reg += d * d;
        }
        out[NSAMP * NLEAF + NSAMP] = 10.0f * reg + bce;   // loss
    }
}

extern "C" void kernel_launch(void* const* d_in, const int* in_sizes, int n_in,
                              void* d_out, int out_size, void* d_ws, size_t ws_size,
                              hipStream_t stream) {
    const float* x      = (const float*)d_in[0];
    const float* y      = (const float*)d_in[1];
    const float* gp     = (const float*)d_in[2];
    const float* refg   = (const float*)d_in[3];
    const int*   d1     = (const int*)  d_in[4];
    const int*   d2     = (const int*)  d_in[5];
    const float* pmat   = (const float*)d_in[6];
    const float* lw     = (const float*)d_in[7];
    const float* lb     = (const float*)d_in[8];
    float* out = (float*)d_out;
    float* acc = (float*)d_ws;            // 64*8 f32 accumulators

    zero_acc_kernel<<<1, NSAMP * NLEAF, 0, stream>>>(acc);
    tree_partial_kernel<<<dim3(NSAMP, CB), TPB, 0, stream>>>(x, gp, d1, d2, pmat, acc);
    finalize_kernel<<<1, NSAMP, 0, stream>>>(acc, y, gp, refg, lw, lb, out);
}